// SS2DTem_39848706573421
// MI455X (gfx1250) — compile-verified
//
#include <hip/hip_runtime.h>
#include <hip/hip_bf16.h>
#include <math.h>

typedef float v2f __attribute__((ext_vector_type(2)));
typedef float v8f __attribute__((ext_vector_type(8)));

#define BATCH   2
#define Hs      56
#define Ws      56
#define Ls      (Hs*Ws)        /* 3136 */
#define DMODEL  96
#define DINNER  192
#define DSTATE  16
#define DTRANK  6
#define NCOUT   38             /* DT_RANK + 2*D_STATE */
#define NKO     (4*NCOUT)      /* 152 */
#define NKO_PAD 160
#define KBIG    (DINNER*2*9)   /* 3456 */
#define THETA_C 0.6f
#define MROWS   (BATCH*Ls)     /* 6272 */

static __device__ __forceinline__ float siluf(float v) { return v / (1.f + expf(-v)); }

static __device__ __forceinline__ v8f wmma4(v2f a, v2f b, v8f c) {
  // D = A(16x4,f32) * B(4x16,f32) + C(16x16,f32), wave32
  return __builtin_amdgcn_wmma_f32_16x16x4_f32(false, a, false, b, (short)0, c, false, false);
}

// ---------------------------------------------------------------------------
// Generic f32 WMMA GEMM:  C[M x N] = A[M x K] * Bt[N x K]^T
// One wave per 16x16 output tile. grid = (N/16, M/16), block = 32.
// ---------------------------------------------------------------------------
__global__ void gemm_f32_wmma(const float* __restrict__ A, const float* __restrict__ Bt,
                              float* __restrict__ C, int K, int ldc) {
  const int lane = threadIdx.x & 31;
  const int half = lane >> 4;
  const int klo  = half << 1;
  const int ln   = lane & 15;
  const int ntile = blockIdx.x;
  const int mtile = blockIdx.y;
  const float* ap = A  + (size_t)(mtile * 16 + ln) * K + klo;
  const float* bp = Bt + (size_t)(ntile * 16 + ln) * K + klo;
  v8f acc = {0.f,0.f,0.f,0.f,0.f,0.f,0.f,0.f};
  for (int k = 0; k < K; k += 4) {
    v2f a = *(const v2f*)(ap + k);
    v2f b = *(const v2f*)(bp + k);
    acc = wmma4(a, b, acc);
  }
  const int n = ntile * 16 + ln;
  for (int rr = 0; rr < 8; ++rr) {
    int mm = mtile * 16 + rr + half * 8;
    C[(size_t)mm * ldc + n] = acc[rr];
  }
}

// ---------------------------------------------------------------------------
// Pack xproj_w (4,38,192,2,3,3) into Wp[160][3456] with kk=((t*3+dy)*3+dx)*192+c
// (zero padded rows 152..159), and kernel_diff kdp[160][192].
// ---------------------------------------------------------------------------
__global__ void k_pack_w(const float* __restrict__ xw, float* __restrict__ Wp) {
  int tid = blockIdx.x * blockDim.x + threadIdx.x;
  if (tid >= NKO_PAD * KBIG) return;
  int kk = tid % KBIG;
  int n  = tid / KBIG;
  float v = 0.f;
  if (n < NKO) {
    int c = kk % DINNER;
    int r = kk / DINNER;
    int dx = r % 3, dy = (r / 3) % 3, t = r / 9;
    v = xw[((((size_t)n * DINNER + c) * 2 + t) * 3 + dy) * 3 + dx];
  }
  Wp[tid] = v;
}

__global__ void k_pack_kd(const float* __restrict__ xw, float* __restrict__ kdp) {
  int tid = blockIdx.x * blockDim.x + threadIdx.x;
  if (tid >= NKO_PAD * DINNER) return;
  int c = tid % DINNER;
  int n = tid / DINNER;
  float v = 0.f;
  if (n < NKO) {
    const float* p = xw + ((size_t)n * DINNER + c) * 18;
    for (int i = 0; i < 18; ++i) v += p[i];
  }
  kdp[tid] = v;
}

// ---------------------------------------------------------------------------
// Depthwise/grouped 3x3 convs + bias + SiLU -> x_in[b][t][h][w][c] (channel-last)
//   t=0: xc from xi (first half of xz, groups=192)
//   t=1: xtc from xt (groups=96 -> channel c reads xt channel c>>1)
// ---------------------------------------------------------------------------
__global__ void k2_conv(const float* __restrict__ xz, const float* __restrict__ xt,
                        const float* __restrict__ w2, const float* __restrict__ b2,
                        const float* __restrict__ wxt, const float* __restrict__ bxt,
                        float* __restrict__ xin) {
  int tid = blockIdx.x * blockDim.x + threadIdx.x;
  if (tid >= BATCH * 2 * Ls * DINNER) return;
  int c = tid % DINNER; int r = tid / DINNER;
  int w = r % Ws; r /= Ws;
  int h = r % Hs; r /= Hs;
  int t = r % 2;  int b = r / 2;
  float acc;
  if (t == 0) {
    acc = b2[c];
    for (int dy = 0; dy < 3; ++dy) {
      int hh = h + dy - 1; if (hh < 0 || hh >= Hs) continue;
      for (int dx = 0; dx < 3; ++dx) {
        int ww = w + dx - 1; if (ww < 0 || ww >= Ws) continue;
        acc += xz[((size_t)((b * Hs + hh) * Ws + ww)) * 384 + c] * w2[c * 9 + dy * 3 + dx];
      }
    }
  } else {
    acc = bxt[c];
    int ci = c >> 1;
    for (int dy = 0; dy < 3; ++dy) {
      int hh = h + dy - 1; if (hh < 0 || hh >= Hs) continue;
      for (int dx = 0; dx < 3; ++dx) {
        int ww = w + dx - 1; if (ww < 0 || ww >= Ws) continue;
        acc += xt[((size_t)((b * Hs + hh) * Ws + ww)) * DMODEL + ci] * wxt[c * 9 + dy * 3 + dx];
      }
    }
  }
  xin[tid] = siluf(acc);
}

// ---------------------------------------------------------------------------
// Pooling + cross-scan construction: xs[B][4][192][L]
//   k0: inter_row at (h=l%H, w=l/H);  k1: same with h flipped
//   k2: inter_col at (h=l/W, w=l%W);  k3: same with w flipped
// ---------------------------------------------------------------------------
__global__ void k3_xs(const float* __restrict__ xin, float* __restrict__ xs) {
  int tid = blockIdx.x * blockDim.x + threadIdx.x;
  if (tid >= BATCH * DINNER * Ls) return;
  int l = tid % Ls; int r = tid / Ls;
  int d = r % DINNER; int b = r / DINNER;

  auto X = [&](int s, int hh, int ww) -> float {
    return xin[(((size_t)(b * 2 + s) * Hs + hh) * Ws + ww) * DINNER + d];
  };
  auto irow = [&](int hh, int ww) -> float {
    int s = hh & 1, i = hh >> 1;
    float v0 = X(s, 2 * i, ww), v1 = X(s, 2 * i + 1, ww);
    return 0.5f * (v0 + v1) + fmaxf(v0, v1);
  };
  auto icol = [&](int hh, int ww) -> float {
    int s = ww & 1, j = ww >> 1;
    float v0 = X(s, hh, 2 * j), v1 = X(s, hh, 2 * j + 1);
    return 0.5f * (v0 + v1) + fmaxf(v0, v1);
  };

  int wr = l / Hs, hr = l % Hs;   // row-major scan coordinates (l = w*H + h)
  int hc = l / Ws, wc = l % Ws;   // col-major scan coordinates (l = h*W + w)
  size_t stride = (size_t)DINNER * Ls;
  size_t base = ((size_t)(b * 4) * DINNER + d) * Ls + l;
  xs[base + 0 * stride] = irow(hr, wr);
  xs[base + 1 * stride] = irow(Hs - 1 - hr, wr);
  xs[base + 2 * stride] = icol(hc, wc);
  xs[base + 3 * stride] = icol(hc, Ws - 1 - wc);
}

// ---------------------------------------------------------------------------
// Fused CDC: conv3d implicit GEMM (K=3456) + two K=192 diff GEMMs + t-pooling.
// One wave per 16x16 tile; grid = (10, 392), block = 32.
// pooled = 2*N - theta*( (D0+D1)/2 + min(D0,D1) )
// ---------------------------------------------------------------------------
__global__ void k4_cdc(const float* __restrict__ xin, const float* __restrict__ Wp,
                       const float* __restrict__ kdp, float* __restrict__ xdbl) {
  const int lane = threadIdx.x & 31;
  const int half = lane >> 4;
  const int klo  = half << 1;
  const int ln   = lane & 15;
  const int ntile = blockIdx.x;
  const int mtile = blockIdx.y;
  const int m = mtile * 16 + ln;
  const int b = m / Ls;
  const int hw = m % Ls;
  const int h = hw / Ws, w = hw % Ws;
  const int n = ntile * 16 + ln;

  v8f accN  = {0.f,0.f,0.f,0.f,0.f,0.f,0.f,0.f};
  const float* wptr = Wp + (size_t)n * KBIG + klo;
  for (int td = 0; td < 18; ++td) {
    int t = td / 9, rr9 = td % 9, dy = rr9 / 3, dx = rr9 % 3;
    int hh = h + dy - 1, ww = w + dx - 1;
    bool ok = (hh >= 0) && (hh < Hs) && (ww >= 0) && (ww < Ws);
    const float* ap = xin + (((size_t)(b * 2 + t) * Hs + hh) * Ws + ww) * DINNER + klo;
    for (int c = 0; c < DINNER; c += 4) {
      v2f a = {0.f, 0.f};
      if (ok) a = *(const v2f*)(ap + c);
      v2f bb = *(const v2f*)wptr;
      wptr += 4;
      accN = wmma4(a, bb, accN);
    }
  }

  v8f accD0 = {0.f,0.f,0.f,0.f,0.f,0.f,0.f,0.f};
  v8f accD1 = {0.f,0.f,0.f,0.f,0.f,0.f,0.f,0.f};
  const float* ap0 = xin + (((size_t)(b * 2 + 0) * Hs + h) * Ws + w) * DINNER + klo;
  const float* ap1 = xin + (((size_t)(b * 2 + 1) * Hs + h) * Ws + w) * DINNER + klo;
  const float* kp  = kdp + (size_t)n * DINNER + klo;
  for (int c = 0; c < DINNER; c += 4) {
    v2f bb = *(const v2f*)(kp + c);
    v2f a0 = *(const v2f*)(ap0 + c);
    v2f a1 = *(const v2f*)(ap1 + c);
    accD0 = wmma4(a0, bb, accD0);
    accD1 = wmma4(a1, bb, accD1);
  }

  if (n < NKO) {
    int kq = n / NCOUT, o = n % NCOUT;
    for (int rr = 0; rr < 8; ++rr) {
      int mm = mtile * 16 + rr + half * 8;
      int b2 = mm / Ls; int l2 = mm % Ls;
      float Nv = accN[rr], d0 = accD0[rr], d1 = accD1[rr];
      float pooled = 2.f * Nv - THETA_C * (0.5f * (d0 + d1) + fminf(d0, d1));
      xdbl[(((size_t)b2 * 4 + kq) * NCOUT + o) * Ls + l2] = pooled;
    }
  }
}

// ---------------------------------------------------------------------------
// delta = softplus(dt_w @ dts + dt_b),  delta[B][4][192][L]
// ---------------------------------------------------------------------------
__global__ void k5_delta(const float* __restrict__ xdbl, const float* __restrict__ dtw,
                         const float* __restrict__ dtb, float* __restrict__ delta) {
  int tid = blockIdx.x * blockDim.x + threadIdx.x;
  if (tid >= BATCH * 4 * DINNER * Ls) return;
  int l = tid % Ls; int r = tid / Ls;
  int d = r % DINNER; int bk = r / DINNER;
  int k = bk & 3;
  float acc = dtb[k * DINNER + d];
  const float* wp = dtw + ((size_t)k * DINNER + d) * DTRANK;
  const float* xp = xdbl + (size_t)bk * NCOUT * Ls + l;
  for (int rr = 0; rr < DTRANK; ++rr) acc += xp[(size_t)rr * Ls] * wp[rr];
  delta[tid] = fmaxf(acc, 0.f) + log1pf(expf(-fabsf(acc)));   // stable softplus
}

// ---------------------------------------------------------------------------
// Selective scan. One wave handles 2 d-channels x 16 states (lane = half*16+n).
// 768 waves total. Writes out_y = y + Ds*u IN PLACE over delta (safe: each
// delta[l] is consumed by this wave in the same iteration before the store).
// ---------------------------------------------------------------------------
__global__ void k6_scan(const float* __restrict__ xs, const float* __restrict__ xdbl,
                        const float* __restrict__ Alogs, const float* __restrict__ Dss,
                        float* __restrict__ dio) {
  int wid  = blockIdx.x * (blockDim.x >> 5) + (threadIdx.x >> 5);
  int lane = threadIdx.x & 31;
  int half = lane >> 4, ln = lane & 15;
  int bk = wid / 96;
  int d  = (wid % 96) * 2 + half;
  int k  = bk & 3;
  const float* dP = dio  + ((size_t)bk * DINNER + d) * Ls;
  float*       oP = dio  + ((size_t)bk * DINNER + d) * Ls;
  const float* uP = xs   + ((size_t)bk * DINNER + d) * Ls;
  const float* bP = xdbl + ((size_t)bk * NCOUT + DTRANK + ln) * Ls;
  const float* cP = xdbl + ((size_t)bk * NCOUT + DTRANK + DSTATE + ln) * Ls;
  float A  = -expf(Alogs[((size_t)(k * DINNER + d)) * DSTATE + ln]);
  float Dv = Dss[k * DINNER + d];
  float h = 0.f;
  for (int l = 0; l < Ls; ++l) {
    float dl = dP[l];
    float ul = uP[l];
    float bl = bP[l];
    float cl = cP[l];
    h = expf(dl * A) * h + (dl * ul) * bl;
    float y = h * cl;
    y += __shfl_xor(y, 1);
    y += __shfl_xor(y, 2);
    y += __shfl_xor(y, 4);
    y += __shfl_xor(y, 8);
    if (ln == 0) oP[l] = y + Dv * ul;
  }
}

// ---------------------------------------------------------------------------
// Cross-merge of the 4 directions -> y2[m][d], m = b*L + (h*W+w)
// ---------------------------------------------------------------------------
__global__ void k7a_comb(const float* __restrict__ oy, float* __restrict__ y2) {
  int tid = blockIdx.x * blockDim.x + threadIdx.x;
  if (tid >= MROWS * DINNER) return;
  int d = tid % DINNER; int m = tid / DINNER;
  int b = m / Ls; int l = m % Ls;
  int h = l / Ws, w = l % Ws;
  int lt = w * Hs + h;
  size_t s = (size_t)DINNER * Ls;
  const float* base = oy + ((size_t)b * 4 * DINNER + d) * Ls;
  y2[tid] = base[l] + base[2 * s + (Ls - 1 - l)]
          + base[s + lt] + base[3 * s + (Ls - 1 - lt)];
}

// ---------------------------------------------------------------------------
// LayerNorm over d=192 per row + SiLU(z) gate (in place on y2).
// block = 192 threads (6 waves), shfl+LDS reduction.
// ---------------------------------------------------------------------------
__global__ void k7b_ln(const float* __restrict__ xz, const float* __restrict__ g,
                       const float* __restrict__ bb, float* __restrict__ y2) {
  int m = blockIdx.x;
  int d = threadIdx.x;
  float v = y2[(size_t)m * DINNER + d];
  float s1 = v, s2 = v * v;
  #pragma unroll
  for (int off = 16; off >= 1; off >>= 1) {
    s1 += __shfl_xor(s1, off);
    s2 += __shfl_xor(s2, off);
  }
  __shared__ float r1[6], r2[6];
  __shared__ float smu, srs;
  if ((d & 31) == 0) { r1[d >> 5] = s1; r2[d >> 5] = s2; }
  __syncthreads();
  if (d == 0) {
    float t1 = 0.f, t2 = 0.f;
    for (int i = 0; i < 6; ++i) { t1 += r1[i]; t2 += r2[i]; }
    float mu = t1 / (float)DINNER;
    float var = t2 / (float)DINNER - mu * mu;
    smu = mu; srs = rsqrtf(var + 1e-5f);
  }
  __syncthreads();
  float yln = (v - smu) * srs * g[d] + bb[d];
  float zv = xz[(size_t)m * 384 + DINNER + d];
  y2[(size_t)m * DINNER + d] = yln * siluf(zv);
}

// ---------------------------------------------------------------------------
extern "C" void kernel_launch(void* const* d_in, const int* in_sizes, int n_in,
                              void* d_out, int out_size, void* d_ws, size_t ws_size,
                              hipStream_t stream) {
  (void)in_sizes; (void)n_in; (void)out_size; (void)ws_size;
  const float* x   = (const float*)d_in[0];
  const float* xt  = (const float*)d_in[1];
  const float* ipw = (const float*)d_in[2];
  const float* c2w = (const float*)d_in[3];
  const float* c2b = (const float*)d_in[4];
  const float* cxw = (const float*)d_in[5];
  const float* cxb = (const float*)d_in[6];
  const float* xpw = (const float*)d_in[7];
  const float* dtw = (const float*)d_in[8];
  const float* dtb = (const float*)d_in[9];
  const float* alg = (const float*)d_in[10];
  const float* dss = (const float*)d_in[11];
  const float* lng = (const float*)d_in[12];
  const float* lnb = (const float*)d_in[13];
  const float* opw = (const float*)d_in[14];
  float* out = (float*)d_out;
  float* ws  = (float*)d_ws;

  size_t off = 0;
  float* xz    = ws + off; off += (size_t)MROWS * 384;            // 2,408,448
  float* xin   = ws + off; off += (size_t)BATCH * 2 * Ls * DINNER;// 2,408,448
  float* xs    = ws + off; off += (size_t)BATCH * 4 * DINNER * Ls;// 4,816,896
  float* Wp    = ws + off; off += (size_t)NKO_PAD * KBIG;         //   552,960
  float* kdp   = ws + off; off += (size_t)NKO_PAD * DINNER;       //    30,720
  float* xdbl  = ws + off; off += (size_t)BATCH * 4 * NCOUT * Ls; //   953,344
  float* delta = ws + off; off += (size_t)BATCH * 4 * DINNER * Ls;// 4,816,896
  float* y2    = xin;  // xin dead after k4_cdc; reuse for merged y
  // total ~16M floats = ~64 MB

  // 1) in_proj GEMM (WMMA f32): xz[6272,384] = x[6272,96] * ipw[384,96]^T
  gemm_f32_wmma<<<dim3(384 / 16, MROWS / 16), 32, 0, stream>>>(x, ipw, xz, DMODEL, 384);

  // 2) pack conv3d weights + kernel_diff
  k_pack_w <<<(NKO_PAD * KBIG + 255) / 256, 256, 0, stream>>>(xpw, Wp);
  k_pack_kd<<<(NKO_PAD * DINNER + 255) / 256, 256, 0, stream>>>(xpw, kdp);

  // 3) depthwise/grouped convs + SiLU -> x_in (channel-last)
  k2_conv<<<(BATCH * 2 * Ls * DINNER + 255) / 256, 256, 0, stream>>>(
      xz, xt, c2w, c2b, cxw, cxb, xin);

  // 4) pooling + cross-scan -> xs
  k3_xs<<<(BATCH * DINNER * Ls + 255) / 256, 256, 0, stream>>>(xin, xs);

  // 5) fused CDC implicit-GEMM (WMMA f32) + t-pooling -> x_dbl
  k4_cdc<<<dim3(NKO_PAD / 16, MROWS / 16), 32, 0, stream>>>(xin, Wp, kdp, xdbl);

  // 6) dt projection + softplus -> delta
  k5_delta<<<(BATCH * 4 * DINNER * Ls + 255) / 256, 256, 0, stream>>>(xdbl, dtw, dtb, delta);

  // 7) selective scan (768 waves), out_y written in place over delta
  k6_scan<<<96, 256, 0, stream>>>(xs, xdbl, alg, dss, delta);

  // 8) cross-merge -> y2, layernorm+gate in place
  k7a_comb<<<(MROWS * DINNER + 255) / 256, 256, 0, stream>>>(delta, y2);
  k7b_ln<<<MROWS, DINNER, 0, stream>>>(xz, lng, lnb, y2);

  // 9) out_proj GEMM (WMMA f32) -> d_out
  gemm_f32_wmma<<<dim3(DMODEL / 16, MROWS / 16), 32, 0, stream>>>(y2, opw, out, DINNER, DMODEL);
}